// MultiLayerAttention_90391881711712
// MI455X (gfx1250) — compile-verified
//
#include <hip/hip_runtime.h>

#define DIM   256
#define HEADS 8

typedef __attribute__((ext_vector_type(2))) float v2f;
typedef __attribute__((ext_vector_type(8))) float v8f;

// ---------------------------------------------------------------- utilities
__global__ void zero_kernel(float* __restrict__ p, long n) {
    long i = (long)blockIdx.x * blockDim.x + threadIdx.x;
    long stride = (long)gridDim.x * blockDim.x;
    for (; i < n; i += stride) p[i] = 0.0f;
}

// softmax of hop_weights (2) and gate_vector (8); tiny, one thread
__global__ void prep_scalars(const float* __restrict__ hop_in,
                             const float* __restrict__ gate_in,
                             float* __restrict__ hw, float* __restrict__ gates) {
    if (threadIdx.x == 0) {
        float m = fmaxf(hop_in[0], hop_in[1]);
        float e0 = __expf(hop_in[0] - m), e1 = __expf(hop_in[1] - m);
        float s = e0 + e1;
        hw[0] = e0 / s; hw[1] = e1 / s;

        float mg = gate_in[0];
        for (int h = 1; h < HEADS; ++h) mg = fmaxf(mg, gate_in[h]);
        float eg[HEADS]; float sg = 0.0f;
        for (int h = 0; h < HEADS; ++h) { eg[h] = __expf(gate_in[h] - mg); sg += eg[h]; }
        for (int h = 0; h < HEADS; ++h) gates[h] = eg[h] / sg;
    }
}

// W̄ = (1/H) Σ_h Wh_w[h],  b̄ = (1/H) Σ_h Wh_b[h]
__global__ void reduce_wh(const float* __restrict__ Whw, const float* __restrict__ Whb,
                          float* __restrict__ Wsum, float* __restrict__ bsum) {
    int i = blockIdx.x * blockDim.x + threadIdx.x;   // 65536 threads
    float s = 0.0f;
    for (int h = 0; h < HEADS; ++h) s += Whw[(long)h * DIM * DIM + i];
    Wsum[i] = s * (1.0f / HEADS);
    if (i < DIM) {
        float b = 0.0f;
        for (int h = 0; h < HEADS; ++h) b += Whb[h * DIM + i];
        bsum[i] = b * (1.0f / HEADS);
    }
}

// fold gates into projection:  Pw[o][d] = Wp_w[o][d]*g[o/32],  Pb[o] = Wp_b[o]*g[o/32]
__global__ void prep_proj(const float* __restrict__ Wpw, const float* __restrict__ Wpb,
                          const float* __restrict__ gates,
                          float* __restrict__ Pw, float* __restrict__ Pb) {
    int i = blockIdx.x * blockDim.x + threadIdx.x;   // 65536 threads
    int h = i >> 13;                                 // / (32*256)
    Pw[i] = Wpw[i] * gates[h];
    if (i < DIM) Pb[i] = Wpb[i] * gates[i >> 5];
}

// in-degree (float) via atomics
__global__ void degree_kernel(const int* __restrict__ dst, float* __restrict__ deg, int nedges) {
    int e = blockIdx.x * blockDim.x + threadIdx.x;
    if (e < nedges) atomicAdd(&deg[dst[e]], 1.0f);
}

__global__ void invdeg_kernel(const float* __restrict__ deg, float* __restrict__ inv, int n) {
    int i = blockIdx.x * blockDim.x + threadIdx.x;
    if (i < n) { float d = deg[i]; inv[i] = (d > 0.0f) ? (1.0f / d) : 0.0f; }
}

// segment-sum: out[dst] += ht[src]; one block per edge, one lane per feature
__global__ void scatter_edges(const float* __restrict__ ht, const int* __restrict__ src,
                              const int* __restrict__ dst, float* __restrict__ out) {
    int e = blockIdx.x;
    int s = src[e], d = dst[e];
    int t = threadIdx.x;                    // blockDim.x == DIM
    atomicAdd(&out[(long)d * DIM + t], ht[(long)s * DIM + t]);
}

// ---------------------------------------------------------------- WMMA GEMM
// Y[n][o] = (rowScale[n]) * (w0*X1[n,:] + w1*X2[n,:]) · W[o,:]  +  bias[o]
// W is DIM x DIM row-major ([o][d]);  block = 8 waves -> 32x64 output tile,
// K staged through LDS in 64-wide chunks, one 16x16 WMMA tile per wave.
#define BLK_ROWS  32
#define BLK_COLS  64
#define KCHUNK    64
#define LDSS      68   // padded stride: 68*n mod 64 = 4n  -> conflict-free

__global__ __launch_bounds__(256) void gemm_wmma(
        const float* __restrict__ X1, const float* __restrict__ X2,
        const float* __restrict__ rowScale, const float* __restrict__ hopw,
        const float* __restrict__ W, const float* __restrict__ bias,
        float* __restrict__ Y, int nrows)
{
    __shared__ float Xs[BLK_ROWS * LDSS];
    __shared__ float Ws[BLK_COLS * LDSS];

    const int tid  = threadIdx.x;
    const int lane = tid & 31;
    const int wave = tid >> 5;
    const int rowTile = wave >> 2;                 // 0..1
    const int colTile = wave & 3;                  // 0..3
    const int blockRow0 = blockIdx.x * BLK_ROWS;
    const int blockCol0 = blockIdx.y * BLK_COLS;

    const float w0 = hopw ? hopw[0] : 1.0f;
    const float w1 = hopw ? hopw[1] : 0.0f;

    const int mn   = lane & 15;                    // M (A) / N (B,C,D) index
    const int koff = (lane >> 4) << 1;             // lanes 16-31 hold K+2,K+3

    // accumulator initialized with bias (bias depends only on output column)
    v8f acc;
    {
        float b = bias[blockCol0 + colTile * 16 + mn];
        for (int j = 0; j < 8; ++j) acc[j] = b;
    }

    for (int kc = 0; kc < DIM; kc += KCHUNK) {
        // stage X chunk (BLK_ROWS x KCHUNK), with fused hop-mix + row scaling
        for (int idx = tid; idx < BLK_ROWS * KCHUNK; idx += 256) {
            int r = idx >> 6, c = idx & 63;
            int grow = blockRow0 + r;
            float v = 0.0f;
            if (grow < nrows) {
                long g = (long)grow * DIM + kc + c;
                v = w0 * X1[g];
                if (X2) v += w1 * X2[g];
                if (rowScale) v *= rowScale[grow];
            }
            Xs[r * LDSS + c] = v;
        }
        // stage W chunk (BLK_COLS rows of W, KCHUNK cols)
        for (int idx = tid; idx < BLK_COLS * KCHUNK; idx += 256) {
            int r = idx >> 6, c = idx & 63;
            Ws[r * LDSS + c] = W[(long)(blockCol0 + r) * DIM + kc + c];
        }
        __syncthreads();

        const float* xrow = &Xs[(rowTile * 16 + mn) * LDSS];
        const float* wrow = &Ws[(colTile * 16 + mn) * LDSS];
        #pragma unroll
        for (int kk = 0; kk < KCHUNK; kk += 4) {
            v2f a, b;
            a.x = xrow[kk + koff]; a.y = xrow[kk + koff + 1];
            b.x = wrow[kk + koff]; b.y = wrow[kk + koff + 1];
            // D = A(16x4 f32) x B(4x16 f32) + C   -> v_wmma_f32_16x16x4_f32
            acc = __builtin_amdgcn_wmma_f32_16x16x4_f32(
                      false, a, false, b, (short)0, acc, false, false);
        }
        __syncthreads();
    }

    // store: C/D layout -> row = tile + j + 8*(lane>=16), col = base + (lane&15)
    const int outCol = blockCol0 + colTile * 16 + mn;
    const int rbase  = blockRow0 + rowTile * 16 + ((lane >> 4) << 3);
    #pragma unroll
    for (int j = 0; j < 8; ++j) {
        int row = rbase + j;
        if (row < nrows) Y[(long)row * DIM + outCol] = acc[j];
    }
}

// ---------------------------------------------------------------- launcher
extern "C" void kernel_launch(void* const* d_in, const int* in_sizes, int n_in,
                              void* d_out, int out_size, void* d_ws, size_t ws_size,
                              hipStream_t stream) {
    const float* X    = (const float*)d_in[0];   // node_features (N, D)
    const float* Whw  = (const float*)d_in[1];   // (H, D, D)
    const float* Whb  = (const float*)d_in[2];   // (H, D)
    const float* Wpw  = (const float*)d_in[3];   // (H, 32, D)
    const float* Wpb  = (const float*)d_in[4];   // (H, 32)
    const float* gate = (const float*)d_in[5];   // (H,)
    const float* hop  = (const float*)d_in[6];   // (HOPS,)
    const int*   src  = (const int*)d_in[7];     // (E,)
    const int*   dst  = (const int*)d_in[8];     // (E,)

    const int N = in_sizes[0] / DIM;
    const int E = in_sizes[7];

    float* ws = (float*)d_ws;
    size_t off = 0;
    float* buf0  = ws + off; off += (size_t)N * DIM;   // ht (GEMM output)
    float* buf1  = ws + off; off += (size_t)N * DIM;   // hop-1 segment sum
    float* buf2  = ws + off; off += (size_t)N * DIM;   // hop-2 segment sum
    float* Wsum  = ws + off; off += DIM * DIM;
    float* bsum  = ws + off; off += DIM;
    float* Pw    = ws + off; off += DIM * DIM;
    float* Pb    = ws + off; off += DIM;
    float* deg   = ws + off; off += N;
    float* invd  = ws + off; off += N;
    float* hw    = ws + off; off += 8;
    float* gates = ws + off; off += 8;

    const long nd = (long)N * DIM;

    // deterministic init of every accumulated buffer
    zero_kernel<<<2048, 256, 0, stream>>>(buf1, nd);
    zero_kernel<<<2048, 256, 0, stream>>>(buf2, nd);
    zero_kernel<<<256, 256, 0, stream>>>(deg, N);

    prep_scalars<<<1, 32, 0, stream>>>(hop, gate, hw, gates);
    reduce_wh<<<DIM * DIM / 256, 256, 0, stream>>>(Whw, Whb, Wsum, bsum);
    prep_proj<<<DIM * DIM / 256, 256, 0, stream>>>(Wpw, Wpb, gates, Pw, Pb);

    degree_kernel<<<(E + 255) / 256, 256, 0, stream>>>(dst, deg, E);
    invdeg_kernel<<<(N + 255) / 256, 256, 0, stream>>>(deg, invd, N);

    dim3 ggrid((N + BLK_ROWS - 1) / BLK_ROWS, DIM / BLK_COLS);

    // hop 1: ht = X @ W̄ᵀ + b̄ ; scatter-sum into buf1
    gemm_wmma<<<ggrid, 256, 0, stream>>>(X, nullptr, nullptr, nullptr,
                                         Wsum, bsum, buf0, N);
    scatter_edges<<<E, DIM, 0, stream>>>(buf0, src, dst, buf1);

    // hop 2: ht = (buf1 * invdeg) @ W̄ᵀ + b̄ ; scatter-sum into buf2
    gemm_wmma<<<ggrid, 256, 0, stream>>>(buf1, nullptr, invd, nullptr,
                                         Wsum, bsum, buf0, N);
    scatter_edges<<<E, DIM, 0, stream>>>(buf0, src, dst, buf2);

    // final: out = [(hw0*buf1 + hw1*buf2) * invdeg] @ Pwᵀ + Pb   (gates folded in)
    gemm_wmma<<<ggrid, 256, 0, stream>>>(buf1, buf2, invd, hw,
                                         Pw, Pb, (float*)d_out, N);
}